// class_aware_rgat_11622181503321
// MI455X (gfx1250) — compile-verified
//
#include <hip/hip_runtime.h>
#include <hip/hip_bf16.h>

#define V_SIZE 50000
#define NREL   16
#define DIM    128
#define LSEG   8

#define THREADS 256           // 8 waves (wave32)
#define SEGS_PER_BLOCK 128    // 16 segments per wave

typedef __attribute__((ext_vector_type(2))) float v2f;
typedef __attribute__((ext_vector_type(8))) float v8f;

// Kernel A: invert the permutation; write the zero row (the one sourcing
// all_emb[V]) directly to the output.
__global__ __launch_bounds__(256)
void build_inv_kernel(const int* __restrict__ pi, int* __restrict__ inv,
                      float* __restrict__ out) {
    int i = blockIdx.x * 256 + threadIdx.x;
    if (i < V_SIZE + 1) {
        int p = pi[i];
        inv[p] = i;
        if (p == V_SIZE) {
            float4* o = (float4*)out + (size_t)i * (DIM / 4);
            float4 z; z.x = 0.f; z.y = 0.f; z.z = 0.f; z.w = 0.f;
            #pragma unroll
            for (int k = 0; k < DIM / 4; ++k) o[k] = z;
        }
    }
}

// Kernel B: fused class-coefficient GEMV (WMMA f32 16x16x4) + gather-sum
// over L ancestors + permuted scatter of finished rows.
__global__ __launch_bounds__(THREADS)
void rgat_main_kernel(const float* __restrict__ classEmb,
                      const float* __restrict__ reEmb,
                      const float* __restrict__ anEmbTab,
                      const float* __restrict__ Wc,
                      const float* __restrict__ bc,
                      const int*   __restrict__ leaves,
                      const int*   __restrict__ ancestors,
                      const int*   __restrict__ relations,
                      const int*   __restrict__ inv,
                      float*       __restrict__ out) {
    __shared__ float s_re[(NREL + 1) * DIM];   // 8704 B: whole reEmb table
    __shared__ float s_coe[SEGS_PER_BLOCK];

    const int tid    = threadIdx.x;
    const int lane   = tid & 31;
    const int wave   = tid >> 5;
    const int vblock = blockIdx.x * SEGS_PER_BLOCK;

    // reEmb table -> LDS, float4 granularity (544 float4s over 256 threads)
    {
        float4*       s4 = (float4*)s_re;
        const float4* g4 = (const float4*)reEmb;
        for (int i = tid; i < (NREL + 1) * DIM / 4; i += THREADS) s4[i] = g4[i];
    }
    __syncthreads();

    const float bcv = bc[0];

    // ---- Phase 1: class_coe for this wave's 16 segments via WMMA ----
    // D(16x16) = A(16x4) x B(4x16) accumulated over 32 K-chunks (K=128).
    // A: lanes 0-15 hold rows M with K=0,1 in v0,v1; lanes 16-31 hold K=2,3.
    // B: the SAME Wc chunk is broadcast into every column, so every column
    //    of D carries the GEMV result; we read column 0 via lanes 0/16.
    {
        const int  m   = lane & 15;
        const int  hi  = lane >> 4;
        const bool hib = (lane >= 16);        // loop-invariant lane predicate
        int v   = vblock + wave * 16 + m;
        int vc  = (v < V_SIZE) ? v : 0;
        int row = leaves[vc * LSEG];          // leaves[:,0] gather
        const float2* arow =
            (const float2*)(classEmb + (size_t)row * DIM + 2 * hi);
        const float4* wc4 = (const float4*)Wc;  // wave-uniform -> s_load_b128

        v8f acc = {};
        #pragma unroll
        for (int s = 0; s < 32; ++s) {
            float2 av = arow[2 * s];          // per-lane A fragment (b64 load)
            float4 w  = wc4[s];               // scalar-path Wc chunk
            v2f a; a.x = av.x; a.y = av.y;
            v2f b;                            // broadcast chunk: 2 cndmasks
            b.x = hib ? w.z : w.x;
            b.y = hib ? w.w : w.y;
            acc = __builtin_amdgcn_wmma_f32_16x16x4_f32(
                false, a, false, b, (short)0, acc, false, false);
        }
        // Column 0 of D: lane 0 holds M=0..7 in v0..v7; lane 16 holds M=8..15.
        if (m == 0) {
            #pragma unroll
            for (int r = 0; r < 8; ++r)
                s_coe[wave * 16 + hi * 8 + r] = acc[r] + bcv;
        }
    }
    __syncthreads();

    // ---- Phase 2: temp[v,:] = sum_l anc + coe * rel ; permuted scatter ----
    for (int t = 0; t < 16; ++t) {
        int v = vblock + wave * 16 + t;          // uniform across the wave
        if (v >= V_SIZE) break;
        int vu = __builtin_amdgcn_readfirstlane(v);   // force SGPR

        const int* ap = ancestors + (size_t)vu * LSEG;  // uniform -> s_load
        const int* rp = relations + (size_t)vu * LSEG;  // uniform -> s_load
        const float c = s_coe[wave * 16 + t];

        float ax = 0.f, ay = 0.f, az = 0.f, aw = 0.f;
        #pragma unroll
        for (int l = 0; l < LSEG; ++l) {
            int ai = ap[l];                       // SGPR index
            int ri = rp[l];                       // SGPR index
            float4 a  = *((const float4*)(anEmbTab + (size_t)ai * DIM) + lane);
            float4 re = *((const float4*)(s_re + ri * DIM) + lane);
            ax += a.x + c * re.x;
            ay += a.y + c * re.y;
            az += a.z + c * re.z;
            aw += a.w + c * re.w;
        }
        int ou = inv[vu];                         // uniform -> s_load
        float4 res; res.x = ax; res.y = ay; res.z = az; res.w = aw;
        *((float4*)out + (size_t)ou * (DIM / 4) + lane) = res;
    }
}

extern "C" void kernel_launch(void* const* d_in, const int* in_sizes, int n_in,
                              void* d_out, int out_size, void* d_ws, size_t ws_size,
                              hipStream_t stream) {
    // setup_inputs order:
    // 0 dxEmb (dead) 1 classEmb 2 reEmb 3 anEmbTab 4 W1(dead) 5 b1(dead)
    // 6 W2(dead) 7 b2(dead) 8 Wc 9 bc 10 leaves 11 ancestors 12 relations
    // 13 permute_index
    const float* classEmb  = (const float*)d_in[1];
    const float* reEmb     = (const float*)d_in[2];
    const float* anEmbTab  = (const float*)d_in[3];
    const float* Wc        = (const float*)d_in[8];
    const float* bc        = (const float*)d_in[9];
    const int*   leaves    = (const int*)d_in[10];
    const int*   ancestors = (const int*)d_in[11];
    const int*   relations = (const int*)d_in[12];
    const int*   pi        = (const int*)d_in[13];
    float*       out       = (float*)d_out;
    int*         inv       = (int*)d_ws;    // (V+1) ints = 200 KB scratch

    build_inv_kernel<<<(V_SIZE + 1 + 255) / 256, 256, 0, stream>>>(pi, inv, out);

    int nblocks = (V_SIZE + SEGS_PER_BLOCK - 1) / SEGS_PER_BLOCK;
    rgat_main_kernel<<<nblocks, THREADS, 0, stream>>>(
        classEmb, reEmb, anEmbTab, Wc, bc, leaves, ancestors, relations, inv, out);
}